// TorchHOGMulti_1700807049340
// MI455X (gfx1250) — compile-verified
//
#include <hip/hip_runtime.h>
#include <stdint.h>

#define NTHREADS 256

// ---------------------------------------------------------------------------
// Zero-padded 3x3 neighborhood fetch from the LDS image tile.
// ---------------------------------------------------------------------------
__device__ __forceinline__ float pix(const float* img, int y, int x) {
  bool oob = ((unsigned)y >= 28u) || ((unsigned)x >= 28u);
  int yy = y < 0 ? 0 : (y > 27 ? 27 : y);
  int xx = x < 0 ? 0 : (x > 27 ? 27 : x);
  float v = img[yy * 28 + xx];
  return oob ? 0.0f : v;
}

// ---------------------------------------------------------------------------
// Soft-assign one pixel's gradient into a cell histogram (LDS float atomics).
// All divisors are compile-time constants.
// ---------------------------------------------------------------------------
template <int BINS, int CELL, int HC>
__device__ __forceinline__ void scatter_hog(float ang, float mag, int y, int x,
                                            float* hist) {
  int cy = y / CELL; if (cy > HC - 1) cy = HC - 1;
  int cx = x / CELL; if (cx > HC - 1) cx = HC - 1;
  constexpr float BW = 180.0f / (float)BINS;   // 22.5 / 20 / 30 (exact)
  float a = ang / BW;
  int b0 = (int)floorf(a);
  b0 = b0 < 0 ? 0 : (b0 > BINS - 1 ? BINS - 1 : b0);
  float frac = a - (float)b0;
  int b1 = (b0 == BINS - 1) ? 0 : b0 + 1;
  float* base = hist + (cy * HC + cx) * BINS;
  atomicAdd(base + b0, mag * (1.0f - frac));   // -> ds_add_f32 (no return)
  atomicAdd(base + b1, mag * frac);
}

// ---------------------------------------------------------------------------
// Block gather + L2-hys normalization + standardized coalesced store.
// Pass A: per-block inverse norms into LDS scratch.
// Pass B: stream all NB*NB*4*BINS features with coalesced NT global stores
//         (write-once output; keep it out of WGP$/L2 working set).
// ---------------------------------------------------------------------------
template <int BINS, int HC>
__device__ __forceinline__ void hog_norm_store(
    const float* hist, float* s_inv, float* __restrict__ outp,
    const float* __restrict__ mean, const float* __restrict__ stdv, int tid) {
  constexpr int NB = HC - 1;
  constexpr int BF = 4 * BINS;

  for (int b = tid; b < NB * NB; b += NTHREADS) {
    int by = b / NB, bx = b - by * NB;
    float ss = 0.0f;
#pragma unroll
    for (int q = 0; q < 4; ++q) {
      const float* cp = hist + ((by + (q >> 1)) * HC + (bx + (q & 1))) * BINS;
#pragma unroll
      for (int k = 0; k < BINS; ++k) { float v = cp[k]; ss += v * v; }
    }
    float inv1 = 1.0f / sqrtf(ss + 1e-6f);
    float ss2 = 0.0f;
#pragma unroll
    for (int q = 0; q < 4; ++q) {
      const float* cp = hist + ((by + (q >> 1)) * HC + (bx + (q & 1))) * BINS;
#pragma unroll
      for (int k = 0; k < BINS; ++k) {
        float v = fminf(cp[k] * inv1, 0.2f);
        ss2 += v * v;
      }
    }
    float inv2 = 1.0f / sqrtf(ss2 + 1e-6f);
    s_inv[2 * b]     = inv1;
    s_inv[2 * b + 1] = inv2;
  }
  __syncthreads();

  for (int i = tid; i < NB * NB * BF; i += NTHREADS) {
    int b = i / BF, f = i - b * BF;
    int q = f / BINS, k = f - q * BINS;
    int by = b / NB, bx = b - by * NB;
    float v = hist[((by + (q >> 1)) * HC + (bx + (q & 1))) * BINS + k];
    v = fminf(v * s_inv[2 * b], 0.2f) * s_inv[2 * b + 1];
    __builtin_nontemporal_store((v - mean[i]) / stdv[i], &outp[i]);
  }
  __syncthreads();   // s_inv is reused by the next scale
}

// ---------------------------------------------------------------------------
// One workgroup (8 wave32) per image.
// ---------------------------------------------------------------------------
__global__ __launch_bounds__(NTHREADS) void hog_multi_kernel(
    const float* __restrict__ x, const float* __restrict__ mean,
    const float* __restrict__ stdv, float* __restrict__ out) {
  __shared__ __align__(16) float s_img[784];          // 28x28 tile
  __shared__ float s_hist[392 + 729 + 1176];          // cell4 | cell3 | cell2
  __shared__ float s_inv[2 * 13 * 13];                // per-block norm scratch

  const int tid = threadIdx.x;
  const int img = blockIdx.x;
  const float* src = x + (size_t)img * 784;
  float* outrow = out + (size_t)img * 8296;

  // --- Phase 0: async DMA image -> LDS (ASYNCcnt path), zero hists meanwhile.
  if (tid < 196) {                                    // 196 x b128 = 3136 B
    uint32_t ldsa = (uint32_t)(uintptr_t)(&s_img[0]) + (uint32_t)tid * 16u;
    uint32_t voff = (uint32_t)tid * 16u;
    uint64_t base = (uint64_t)(uintptr_t)src;
    asm volatile("global_load_async_to_lds_b128 %0, %1, %2"
                 :: "v"(ldsa), "v"(voff), "s"(base) : "memory");
  }
  for (int i = tid; i < 392 + 729 + 1176; i += NTHREADS) s_hist[i] = 0.0f;
  asm volatile("s_wait_asynccnt 0" ::: "memory");
  __syncthreads();

  float* s_h4 = s_hist;
  float* s_h3 = s_hist + 392;
  float* s_h2 = s_hist + 392 + 729;

  // --- Phase 1: Sobel + angle + soft-binned scatter into 3 histograms.
  for (int p = tid; p < 784; p += NTHREADS) {
    int y = p / 28, xq = p - y * 28;
    float a00 = pix(s_img, y - 1, xq - 1), a01 = pix(s_img, y - 1, xq),
          a02 = pix(s_img, y - 1, xq + 1);
    float a10 = pix(s_img, y, xq - 1), a12 = pix(s_img, y, xq + 1);
    float a20 = pix(s_img, y + 1, xq - 1), a21 = pix(s_img, y + 1, xq),
          a22 = pix(s_img, y + 1, xq + 1);
    float gx = (a00 + 2.0f * a10 + a20 - a02 - 2.0f * a12 - a22) * 0.25f;
    float gy = (a00 + 2.0f * a01 + a02 - a20 - 2.0f * a21 - a22) * 0.25f;
    float mag = sqrtf(gx * gx + gy * gy + 1e-6f);
    float ang = atan2f(gy, gx) * 57.29577951308232f;  // rad -> deg
    ang = fmodf(ang + 180.0f, 180.0f);                // [0, 180)
    scatter_hog<8, 4, 7>(ang, mag, y, xq, s_h4);
    scatter_hog<9, 3, 9>(ang, mag, y, xq, s_h3);
    scatter_hog<6, 2, 14>(ang, mag, y, xq, s_h2);
  }
  __syncthreads();

  // --- Phase 2a: raw pixel passthrough, standardized, coalesced NT stores.
  for (int i = tid; i < 784; i += NTHREADS)
    __builtin_nontemporal_store((s_img[i] - mean[i]) / stdv[i], &outrow[i]);

  // --- Phase 2b: per-scale block normalization + store.
  hog_norm_store<8, 7>(s_h4, s_inv, outrow + 784,  mean + 784,  stdv + 784,  tid);
  hog_norm_store<9, 9>(s_h3, s_inv, outrow + 1936, mean + 1936, stdv + 1936, tid);
  hog_norm_store<6, 14>(s_h2, s_inv, outrow + 4240, mean + 4240, stdv + 4240, tid);
}

extern "C" void kernel_launch(void* const* d_in, const int* in_sizes, int n_in,
                              void* d_out, int out_size, void* d_ws,
                              size_t ws_size, hipStream_t stream) {
  (void)n_in; (void)out_size; (void)d_ws; (void)ws_size;
  const float* x    = (const float*)d_in[0];
  const float* mean = (const float*)d_in[1];
  const float* stdv = (const float*)d_in[2];
  float* out = (float*)d_out;
  const int B = in_sizes[0] / 784;                    // 16384 images
  hipLaunchKernelGGL(hog_multi_kernel, dim3(B), dim3(NTHREADS), 0, stream,
                     x, mean, stdv, out);
}